// Actor_59133109732210
// MI455X (gfx1250) — compile-verified
//
#include <hip/hip_runtime.h>
#include <math.h>

typedef __attribute__((ext_vector_type(2))) float v2f;
typedef __attribute__((ext_vector_type(8))) float v8f;

#define NB 16384
#define NN 10
#define TT 20
#define BPG 32                 // batch elements per workgroup
#define RPG 320                // rows (b,n) per workgroup
#define NTHREADS 320           // 10 waves
#define OUT_HALF 9830400       // 16384*10*3*20
#define BN_EPS 1e-5f
#define LOG_HALF -0.69314718055994531f

// ---- DPP16 lane permutes (pure VALU, no LDS): butterfly within 16-lane halves
template <int CTRL>
__device__ __forceinline__ float dppf(float x) {
  return __int_as_float(__builtin_amdgcn_update_dpp(
      0, __float_as_int(x), CTRL, 0xF, 0xF, true));
}
template <int CTRL>
__device__ __forceinline__ int dppi(int x) {
  return __builtin_amdgcn_update_dpp(0, x, CTRL, 0xF, 0xF, true);
}
#define DPP_X1 0xB1    // quad_perm(1,0,3,2)  ~ xor 1
#define DPP_X2 0x4E    // quad_perm(2,3,0,1)  ~ xor 2
#define DPP_X4 0x141   // row_half_mirror     ~ xor 4 (quads uniform)
#define DPP_X8 0x140   // row_mirror          ~ xor 8 (8-groups uniform)

// ---------------------------------------------------------------------------
// Fused Actor forward.  Folded linear algebra:
//   logits2[j] = st·Ms[:,j] + dyn·Md[:,j] + C2[j]      (32 cols, via WMMA)
//   g          = st·Gs + dyn·Gd + Gc                    (a3 head collapses to sigmoid)
// WMMA f32 16x16x4: A = [16 rows x K8] (K = st0,st1,st2,dy0,dy1,dy2,0,0),
// B = folded mats, two N-tiles (cols 0-15 / 16-31), bias preloaded into C.
// ---------------------------------------------------------------------------
__global__ __launch_bounds__(NTHREADS) void actor_fused(
    const float* __restrict__ g_static, const float* __restrict__ g_dynamic,
    const float* __restrict__ s_enc_w,  const float* __restrict__ s_enc_b,
    const float* __restrict__ d_enc_w,  const float* __restrict__ d_enc_b,
    const float* __restrict__ s_bn_g,   const float* __restrict__ s_bn_b,
    const float* __restrict__ s_bn_m,   const float* __restrict__ s_bn_v,
    const float* __restrict__ d_bn_g,   const float* __restrict__ d_bn_b,
    const float* __restrict__ d_bn_m,   const float* __restrict__ d_bn_v,
    const float* __restrict__ a2_w,     const float* __restrict__ a2_b,
    const float* __restrict__ a3_w,     const float* __restrict__ a3_b,
    float* __restrict__ out)
{
  __shared__ float fold[232];       // [0,96) Ms  [96,192) Md  [192,224) C2  [224..230] Gs,Gd,Gc
  __shared__ float lds_dyn[RPG*3];  // recurrent state
  __shared__ float lds_pq [RPG];
  __shared__ float lds_lq [RPG];
  __shared__ float lds_bs [RPG];
  __shared__ float lds_cd [RPG];

  const int tid  = threadIdx.x;
  const int lane = tid & 31;
  const int wave = tid >> 5;
  const int col  = lane & 15;       // WMMA column / row-in-tile index
  const int hi   = lane >> 4;       // 0: K0,K1 (K4,K5) half, 1: K2,K3 (K6,K7) half
  const int row0 = blockIdx.x * RPG;

  // ---- fold the whole linear stack into 3x32 mats + 3-vecs (redundant per WG, tiny)
  if (tid < 96) {
    int c = tid >> 5, j = tid & 31;
    float as = s_bn_g[c] * rsqrtf(s_bn_v[c] + BN_EPS);
    float ad = d_bn_g[c] * rsqrtf(d_bn_v[c] + BN_EPS);
    float ps = 0.f, pd = 0.f;
    for (int h = 0; h < 128; ++h) {
      ps += s_enc_w[c*128+h] * a2_w[h*32+j];
      pd += d_enc_w[c*128+h] * a2_w[(128+h)*32+j];
    }
    fold[c*32+j]      = as * ps;
    fold[96 + c*32+j] = ad * pd;
  } else if (tid < 128) {
    int j = tid - 96;
    float acc = a2_b[j];
    for (int c = 0; c < 3; ++c) {
      float as = s_bn_g[c]*rsqrtf(s_bn_v[c]+BN_EPS);
      float bs = s_bn_b[c] - s_bn_m[c]*as;
      float ad = d_bn_g[c]*rsqrtf(d_bn_v[c]+BN_EPS);
      float bd = d_bn_b[c] - d_bn_m[c]*ad;
      float ps = 0.f, pd = 0.f;
      for (int h = 0; h < 128; ++h) {
        ps += s_enc_w[c*128+h]*a2_w[h*32+j];
        pd += d_enc_w[c*128+h]*a2_w[(128+h)*32+j];
      }
      acc += bs*ps + bd*pd;
    }
    for (int h = 0; h < 128; ++h)
      acc += s_enc_b[h]*a2_w[h*32+j] + d_enc_b[h]*a2_w[(128+h)*32+j];
    fold[192+j] = acc;
  } else if (tid == 128) {
    float gc = a3_b[1] - a3_b[0];
    for (int c = 0; c < 3; ++c) {
      float as = s_bn_g[c]*rsqrtf(s_bn_v[c]+BN_EPS);
      float bs = s_bn_b[c]-s_bn_m[c]*as;
      float ad = d_bn_g[c]*rsqrtf(d_bn_v[c]+BN_EPS);
      float bd = d_bn_b[c]-d_bn_m[c]*ad;
      float qs = 0.f, qd = 0.f;
      for (int h = 0; h < 128; ++h) {
        qs += s_enc_w[c*128+h]*(a3_w[h*2+1]-a3_w[h*2]);
        qd += d_enc_w[c*128+h]*(a3_w[(128+h)*2+1]-a3_w[(128+h)*2]);
      }
      fold[224+c] = as*qs;
      fold[227+c] = ad*qd;
      gc += bs*qs + bd*qd;
    }
    for (int h = 0; h < 128; ++h)
      gc += s_enc_b[h]*(a3_w[h*2+1]-a3_w[h*2]) + d_enc_b[h]*(a3_w[(128+h)*2+1]-a3_w[(128+h)*2]);
    fold[230] = gc;
  }

  // ---- init recurrent state: dyn = dynamic[..., 0]
  for (int i = tid; i < RPG*3; i += NTHREADS) {
    int lr = i / 3, c = i % 3;
    lds_dyn[i] = g_dynamic[((size_t)(row0 + lr)*3 + c)*(TT+1)];
  }

  // ---- constant output channels: action[:,:,0,:]=0, logp[:,:,0,:]=log(0.5)
  {
    size_t grow = (size_t)(row0 + tid);
    float* a0 = out + grow*3*TT;
    float* l0 = out + OUT_HALF + grow*3*TT;
    #pragma unroll
    for (int t = 0; t < TT; ++t) { a0[t] = 0.f; l0[t] = LOG_HALF; }
  }
  __syncthreads();

  // ---- wave-resident B operand for V_WMMA_F32_16X16X4_F32 (constant all steps)
  //   B 4x16 layout: VGPR0 = {K0 | K2}, VGPR1 = {K1 | K3} across the two 16-lane halves
  v2f Bc0[2], Bc1[2]; v8f Cb[2];
  #pragma unroll
  for (int nt = 0; nt < 2; ++nt) {
    int cN = col + nt*16;
    Bc0[nt].x = hi ? fold[2*32+cN]      : fold[0*32+cN];   // K2=Ms[2] | K0=Ms[0]
    Bc0[nt].y = hi ? fold[96+0*32+cN]   : fold[1*32+cN];   // K3=Md[0] | K1=Ms[1]
    Bc1[nt].x = hi ? 0.f                : fold[96+1*32+cN];// K6=0     | K4=Md[1]
    Bc1[nt].y = hi ? 0.f                : fold[96+2*32+cN];// K7=0     | K5=Md[2]
    float c2 = fold[192+cN];
    Cb[nt] = (v8f){c2,c2,c2,c2,c2,c2,c2,c2};               // bias preload
  }
  const float Gs0=fold[224], Gs1=fold[225], Gs2=fold[226];
  const float Gd0=fold[227], Gd1=fold[228], Gd2=fold[229], Gc=fold[230];

  const int tileBase = wave * 2;    // 10 waves x 2 tiles = 20 tiles = 320 rows

  for (int t = 0; t < TT; ++t) {
    // ================= phase 1: WMMA logits + row stats =================
    #pragma unroll
    for (int ti = 0; ti < 2; ++ti) {
      const int lrow = (tileBase + ti)*16 + col;
      const size_t sb = ((size_t)(row0 + lrow)*3)*TT + t;
      // unconditional loads (no exec-mask branching before WMMA)
      float gA  = g_static[sb + (hi ? 2*TT : 0)];   // st0 (lo) | st2 (hi)
      float gB  = g_static[sb + TT];                // st1 (used by lo half)
      float dy0 = lds_dyn[lrow*3+0];
      float dy1 = lds_dyn[lrow*3+1];
      float dy2 = lds_dyn[lrow*3+2];
      v2f A0, A1;
      A0.x = gA;                                    // K0=st0 | K2=st2
      A0.y = hi ? dy0 : gB;                         // K1=st1 | K3=dy0
      A1.x = hi ? 0.f : dy1;                        // K4=dy1 | K6=0
      A1.y = hi ? 0.f : dy2;                        // K5=dy2 | K7=0

      v8f accLo = __builtin_amdgcn_wmma_f32_16x16x4_f32(false, A0, false, Bc0[0], (short)0, Cb[0], false, false);
      accLo     = __builtin_amdgcn_wmma_f32_16x16x4_f32(false, A1, false, Bc1[0], (short)0, accLo, false, false);
      v8f accHi = __builtin_amdgcn_wmma_f32_16x16x4_f32(false, A0, false, Bc0[1], (short)0, Cb[1], false, false);
      accHi     = __builtin_amdgcn_wmma_f32_16x16x4_f32(false, A1, false, Bc1[1], (short)0, accHi, false, false);

      // per-row max / logsumexp / argmax via DPP16 butterflies (no LDS traffic)
      #pragma unroll
      for (int v = 0; v < 8; ++v) {
        float x0 = accLo[v], x1 = accHi[v];
        float mx = fmaxf(x0, x1);
        mx = fmaxf(mx, dppf<DPP_X1>(mx));
        mx = fmaxf(mx, dppf<DPP_X2>(mx));
        mx = fmaxf(mx, dppf<DPP_X4>(mx));
        mx = fmaxf(mx, dppf<DPP_X8>(mx));
        float s = __expf(x0 - mx) + __expf(x1 - mx);
        s += dppf<DPP_X1>(s);
        s += dppf<DPP_X2>(s);
        s += dppf<DPP_X4>(s);
        s += dppf<DPP_X8>(s);
        int i0 = (x0 == mx) ? col        : 1000;
        int i1 = (x1 == mx) ? (col + 16) : 1000;
        int idx = min(i0, i1);
        idx = min(idx, dppi<DPP_X1>(idx));
        idx = min(idx, dppi<DPP_X2>(idx));
        idx = min(idx, dppi<DPP_X4>(idx));
        idx = min(idx, dppi<DPP_X8>(idx));
        if (col == 0) {
          int orow = (tileBase + ti)*16 + v + 8*hi;
          lds_pq[orow] = (float)(idx + 1);      // ptr_quant
          lds_lq[orow] = -__logf(s);            // max(log_softmax) = -log(sum exp(x-mx))
        }
      }
    }
    __syncthreads();

    // ================= phase 2: row-per-thread recurrence =================
    {
      const int lr = tid;
      const int bb = lr / NN;
      const size_t sb2 = ((size_t)(row0 + lr)*3)*TT + t;
      float st0 = g_static[sb2];          // hot in WGP$/L2 (phase 1 touched these lines)
      float st1 = g_static[sb2 + TT];
      float st2 = g_static[sb2 + 2*TT];
      float dy0 = lds_dyn[lr*3], dy1 = lds_dyn[lr*3+1], dy2 = lds_dyn[lr*3+2];
      float g  = st0*Gs0 + st1*Gs1 + st2*Gs2 + dy0*Gd0 + dy1*Gd1 + dy2*Gd2 + Gc;
      float bs = 1.f / (1.f + __expf(-g));      // softmax(l)[1] = sigmoid(l1-l0)
      lds_bs[lr] = bs;
      __syncthreads();

      float ssum = 0.f;
      #pragma unroll
      for (int k = 0; k < NN; ++k) ssum += lds_bs[bb*NN + k];
      float bdw  = bs / ssum;
      float pq   = lds_pq[lr];
      float rate = bdw * 10.f * __log2f(1.f + 1e7f * st0 / (dy2*dy2));
      lds_cd[lr] = 0.002f*pq/st1 + pq/(32.f*rate);
      __syncthreads();

      float d0 = -1e30f;
      #pragma unroll
      for (int k = 0; k < NN; ++k) d0 = fmaxf(d0, lds_cd[bb*NN + k]);
      float d1 = d0*st2 + dy1;
      float dd = d1 + 0.005f*pq - 300.f;
      float d2 = sqrtf(dd*dd + 100.f);
      lds_dyn[lr*3+0] = d0; lds_dyn[lr*3+1] = d1; lds_dyn[lr*3+2] = d2;

      size_t aBase = (size_t)(row0 + lr)*3*TT + t;
      out[aBase + TT]              = pq;         // action ch1
      out[aBase + 2*TT]            = bdw;        // action ch2
      out[OUT_HALF + aBase + TT]   = lds_lq[lr]; // logp  ch1
      out[OUT_HALF + aBase + 2*TT] = bdw;        // logp  ch2
    }
    __syncthreads();   // protects lds_dyn/lds_bs/lds_cd for next step
  }
}

extern "C" void kernel_launch(void* const* d_in, const int* in_sizes, int n_in,
                              void* d_out, int out_size, void* d_ws, size_t ws_size,
                              hipStream_t stream) {
  (void)in_sizes; (void)n_in; (void)out_size; (void)d_ws; (void)ws_size;
  const float* g_static  = (const float*)d_in[0];
  const float* g_dynamic = (const float*)d_in[1];
  actor_fused<<<NB / BPG, NTHREADS, 0, stream>>>(
      g_static, g_dynamic,
      (const float*)d_in[2],  (const float*)d_in[3],
      (const float*)d_in[4],  (const float*)d_in[5],
      (const float*)d_in[6],  (const float*)d_in[7],
      (const float*)d_in[8],  (const float*)d_in[9],
      (const float*)d_in[10], (const float*)d_in[11],
      (const float*)d_in[12], (const float*)d_in[13],
      (const float*)d_in[14], (const float*)d_in[15],
      (const float*)d_in[16], (const float*)d_in[17],
      (float*)d_out);
}